// PointTransformerEncNet_67010079752496
// MI455X (gfx1250) — compile-verified
//
#include <hip/hip_runtime.h>
#include <math.h>

// ---------------------------------------------------------------------------
// Point transformer encoder for MI455X (gfx1250, wave32).
// Heavy GEMMs run on V_WMMA_F32_16X16X4_F32 (exact fp32 accumulate, matches
// the fp32 JAX reference). 4-wave blocks compute 64x64 tiles with LDS-staged,
// double-buffered, coalesced b128 loads; W tile shared by all 4 waves.
// ---------------------------------------------------------------------------

#define BB    32
#define NN    8192
#define GG    128
#define KK    32
#define DIMC  384
#define HEADS_ 6
#define TT    129
#define BT    (BB * TT)      // 4128
#define BG    (BB * GG)      // 4096
#define BGK   (BG * KK)      // 131072

#define KC    32             // k-chunk staged in LDS (divides all K dims used)
#define LDP   36             // padded LDS row stride (floats) -> bank-conflict-free

typedef float v2f __attribute__((ext_vector_type(2)));
typedef float v8f __attribute__((ext_vector_type(8)));

__device__ __forceinline__ float gelu_f(float x) {
  return 0.5f * x * (1.0f + erff(x * 0.70710678118654752440f));
}

// ---------------------------------------------------------------------------
// WMMA f32 GEMM:  out[M,Nc] = act(A' @ W^T + bias) + add
//   A' row r = A[(r / rowdiv)]  (rowdiv>1 broadcasts group features)
//   W is (Nc, wld) row-major; K-loop uses columns [0, Kd) of each W row
//   (pass W+off with wld>Kd to select a column slice).
// Requires Nc%64==0, Kd%32==0. M arbitrary (clamped loads, guarded stores).
// Block = 128 threads (4 waves): 64x64 output tile, wave w -> rows [16w,16w+16).
// WMMA layouts per CDNA5 ISA 7.12.2:
//   A 16x4:  lane m=lane&15, a.x=A[m][kh], a.y=A[m][kh+1], kh=(lane>>4)*2
//   B 4x16:  lane n=lane&15, b.x=B[kh][n], b.y=B[kh+1][n]
//   C 16x16: vgpr r -> row r + 8*(lane>>4), col lane&15
// ---------------------------------------------------------------------------
template<int ACT, bool HASB, bool HASADD>
__global__ __launch_bounds__(128)
void gemm_wmma_kernel(const float* __restrict__ A, const float* __restrict__ W,
                      const float* __restrict__ bias, const float* __restrict__ add,
                      float* __restrict__ out,
                      int M, int Kd, int Nc, int rowdiv, int wld) {
  __shared__ float Al[2][64 * LDP];
  __shared__ float Wl[2][64 * LDP];
  const int tid  = threadIdx.x;
  const int lane = tid & 31;
  const int wv   = tid >> 5;            // wave id 0..3
  const int m0   = blockIdx.x << 6;     // 64 rows per block
  const int n0   = blockIdx.y << 6;     // 64 cols per block
  const int mr   = lane & 15;
  const int kh   = (lane >> 4) << 1;

  v8f acc0 = {}; v8f acc1 = {}; v8f acc2 = {}; v8f acc3 = {};
  float4 ra[4], rw[4];                  // register double-buffer (next chunk)

  auto load_chunk = [&](int kc) {
#pragma unroll
    for (int q = 0; q < 4; ++q) {
      const int f4  = tid + q * 128;    // float4 id in [0,512)
      const int row = f4 >> 3;          // 8 float4 per 32-float row
      const int c4  = f4 & 7;
      int garow = m0 + row; if (garow > M - 1) garow = M - 1;   // clamp (guarded store)
      ra[q] = *(const float4*)(A + (size_t)(garow / rowdiv) * Kd + kc + c4 * 4);
      rw[q] = *(const float4*)(W + (size_t)(n0 + row) * wld + kc + c4 * 4);
    }
  };
  auto store_chunk = [&](int buf) {
#pragma unroll
    for (int q = 0; q < 4; ++q) {
      const int f4  = tid + q * 128;
      const int row = f4 >> 3;
      const int c4  = f4 & 7;
      *(float4*)(&Al[buf][row * LDP + c4 * 4]) = ra[q];
      *(float4*)(&Wl[buf][row * LDP + c4 * 4]) = rw[q];
    }
  };

  load_chunk(0);
  store_chunk(0);
  __syncthreads();

  const int nk = Kd / KC;
  for (int c = 0; c < nk; ++c) {
    if (c + 1 < nk) load_chunk((c + 1) * KC);      // prefetch next chunk (no wait yet)
    const float* Ab = &Al[c & 1][(wv * 16 + mr) * LDP];
    const float* W0 = &Wl[c & 1][mr * LDP];
#pragma unroll
    for (int k = 0; k < KC; k += 4) {
      v2f a;  a.x  = Ab[k + kh];                 a.y  = Ab[k + kh + 1];
      v2f b0; b0.x = W0[k + kh];                 b0.y = W0[k + kh + 1];
      v2f b1; b1.x = W0[16 * LDP + k + kh];      b1.y = W0[16 * LDP + k + kh + 1];
      v2f b2; b2.x = W0[32 * LDP + k + kh];      b2.y = W0[32 * LDP + k + kh + 1];
      v2f b3; b3.x = W0[48 * LDP + k + kh];      b3.y = W0[48 * LDP + k + kh + 1];
      acc0 = __builtin_amdgcn_wmma_f32_16x16x4_f32(false, a, false, b0, (short)0, acc0, false, false);
      acc1 = __builtin_amdgcn_wmma_f32_16x16x4_f32(false, a, false, b1, (short)0, acc1, false, false);
      acc2 = __builtin_amdgcn_wmma_f32_16x16x4_f32(false, a, false, b2, (short)0, acc2, false, false);
      acc3 = __builtin_amdgcn_wmma_f32_16x16x4_f32(false, a, false, b3, (short)0, acc3, false, false);
    }
    __syncthreads();
    if (c + 1 < nk) store_chunk((c + 1) & 1);    // waits for prefetch here
    __syncthreads();
  }

  const int cr = lane & 15;
  const int rb = m0 + wv * 16 + ((lane >> 4) << 3);
#pragma unroll
  for (int r = 0; r < 8; ++r) {
    const int row = rb + r;
    if (row < M) {
      float v[4] = {acc0[r], acc1[r], acc2[r], acc3[r]};
#pragma unroll
      for (int t = 0; t < 4; ++t) {
        const int col = n0 + t * 16 + cr;
        float x = v[t];
        if (HASB)     x += bias[col];
        if (ACT == 1) x = fmaxf(x, 0.0f);
        if (ACT == 2) x = gelu_f(x);
        if (HASADD)   x += add[(size_t)row * Nc + col];
        out[(size_t)row * Nc + col] = x;
      }
    }
  }
}

// ------------------------- FPS (one block per batch) -----------------------
__global__ void fps_kernel(const float* __restrict__ xyz, float* __restrict__ center) {
  const int b = blockIdx.x, tid = threadIdx.x;           // 256 threads
  __shared__ float rval[256]; __shared__ int ridx[256]; __shared__ int sfar;
  const float* xb = xyz + (size_t)b * NN * 3;
  float ds[32];
#pragma unroll
  for (int u = 0; u < 32; ++u) ds[u] = 1e10f;
  int far = 0;
  for (int it = 0; it < GG; ++it) {
    const float cx = xb[far * 3 + 0], cy = xb[far * 3 + 1], cz = xb[far * 3 + 2];
    if (tid == 0) {
      center[((size_t)b * GG + it) * 3 + 0] = cx;
      center[((size_t)b * GG + it) * 3 + 1] = cy;
      center[((size_t)b * GG + it) * 3 + 2] = cz;
    }
    float bv = -1.0f; int bi = 0;
#pragma unroll
    for (int u = 0; u < 32; ++u) {
      const int j = tid * 32 + u;
      const float dx = xb[j * 3 + 0] - cx, dy = xb[j * 3 + 1] - cy, dz = xb[j * 3 + 2] - cz;
      const float nd = fminf(ds[u], dx * dx + dy * dy + dz * dz);
      ds[u] = nd;
      if (nd > bv) { bv = nd; bi = j; }
    }
    rval[tid] = bv; ridx[tid] = bi; __syncthreads();
    for (int st = 128; st > 0; st >>= 1) {
      if (tid < st) {
        const float ov = rval[tid + st]; const int oi = ridx[tid + st];
        if (ov > rval[tid] || (ov == rval[tid] && oi < ridx[tid])) { rval[tid] = ov; ridx[tid] = oi; }
      }
      __syncthreads();
    }
    if (tid == 0) sfar = ridx[0];
    __syncthreads();
    far = sfar;
  }
}

// ------------- KNN top-32 (ascending distance) + gather, per (b,g) ---------
__global__ void knn_kernel(const float* __restrict__ xyz, const float* __restrict__ center,
                           float* __restrict__ neigh) {
  const int bg = blockIdx.x, tid = threadIdx.x;          // 256 threads
  const int b = bg / GG;
  __shared__ float d[NN];
  __shared__ float rval[256]; __shared__ int ridx[256]; __shared__ int ssel;
  const float* xb = xyz + (size_t)b * NN * 3;
  const float cx = center[(size_t)bg * 3 + 0];
  const float cy = center[(size_t)bg * 3 + 1];
  const float cz = center[(size_t)bg * 3 + 2];
  const float cc = cx * cx + cy * cy + cz * cz;
#pragma unroll 4
  for (int u = 0; u < 32; ++u) {
    const int j = tid * 32 + u;
    const float px = xb[j * 3 + 0], py = xb[j * 3 + 1], pz = xb[j * 3 + 2];
    d[j] = cc + (px * px + py * py + pz * pz) - 2.0f * (cx * px + cy * py + cz * pz);
  }
  __syncthreads();
  for (int it = 0; it < KK; ++it) {
    float bv = d[tid * 32]; int bi = tid * 32;
    for (int u = 1; u < 32; ++u) {
      const int j = tid * 32 + u;
      if (d[j] < bv) { bv = d[j]; bi = j; }
    }
    rval[tid] = bv; ridx[tid] = bi; __syncthreads();
    for (int st = 128; st > 0; st >>= 1) {
      if (tid < st) {
        const float ov = rval[tid + st]; const int oi = ridx[tid + st];
        if (ov < rval[tid] || (ov == rval[tid] && oi < ridx[tid])) { rval[tid] = ov; ridx[tid] = oi; }
      }
      __syncthreads();
    }
    if (tid == 0) {
      const int sel = ridx[0];
      ssel = sel;
      neigh[((size_t)bg * KK + it) * 3 + 0] = xb[sel * 3 + 0] - cx;
      neigh[((size_t)bg * KK + it) * 3 + 1] = xb[sel * 3 + 1] - cy;
      neigh[((size_t)bg * KK + it) * 3 + 2] = xb[sel * 3 + 2] - cz;
    }
    __syncthreads();
    if (tid == 0) d[ssel] = 3.0e38f;
    __syncthreads();
  }
}

// ---------------- 3-in -> 128-out linear (conv1 / pos MLP 1) ---------------
template<int ACT>
__global__ void lin3_kernel(const float* __restrict__ in, const float* __restrict__ w,
                            const float* __restrict__ bias, float* __restrict__ out) {
  const int row = blockIdx.x, c = threadIdx.x;           // 128 threads
  const float n0 = in[(size_t)row * 3 + 0];
  const float n1 = in[(size_t)row * 3 + 1];
  const float n2 = in[(size_t)row * 3 + 2];
  float v = n0 * w[c * 3 + 0] + n1 * w[c * 3 + 1] + n2 * w[c * 3 + 2] + bias[c];
  if (ACT == 2) v = gelu_f(v);
  out[(size_t)row * 128 + c] = v;
}

// ----------------- per-column sum / sumsq (for BatchNorm) ------------------
__global__ void colstats_kernel(const float* __restrict__ x, float* __restrict__ stats,
                                int M, int C) {
  const int c = blockIdx.x, tid = threadIdx.x;           // 256 threads
  const int chunk = (M + gridDim.y - 1) / gridDim.y;
  const int r0 = blockIdx.y * chunk;
  const int r1 = (r0 + chunk < M) ? r0 + chunk : M;
  float s = 0.0f, s2 = 0.0f;
  for (int r = r0 + tid; r < r1; r += 256) {
    const float v = x[(size_t)r * C + c];
    s += v; s2 += v * v;
  }
  __shared__ float a1[256], a2[256];
  a1[tid] = s; a2[tid] = s2; __syncthreads();
  for (int st = 128; st > 0; st >>= 1) {
    if (tid < st) { a1[tid] += a1[tid + st]; a2[tid] += a2[tid + st]; }
    __syncthreads();
  }
  if (tid == 0) { atomicAdd(&stats[c], a1[0]); atomicAdd(&stats[C + c], a2[0]); }
}

__global__ void bn_relu_kernel(float* __restrict__ x, const float* __restrict__ stats,
                               const float* __restrict__ g, const float* __restrict__ b,
                               int M, int C) {
  const size_t i = (size_t)blockIdx.x * 256 + threadIdx.x;
  if (i >= (size_t)M * C) return;
  const int c = (int)(i % C);
  const float m = stats[c] / (float)M;
  const float var = stats[C + c] / (float)M - m * m;
  const float v = (x[i] - m) * rsqrtf(var + 1e-5f) * g[c] + b[c];
  x[i] = fmaxf(v, 0.0f);
}

// ----------------------------- group max pools -----------------------------
__global__ void group_max_kernel(const float* __restrict__ f2, float* __restrict__ fg) {
  const int bg = blockIdx.x, c = threadIdx.x;            // 256 threads
  float m = -3.0e38f;
  for (int k = 0; k < KK; ++k) m = fmaxf(m, f2[((size_t)bg * KK + k) * 256 + c]);
  fg[(size_t)bg * 256 + c] = m;
}

__global__ void token_max_kernel(const float* __restrict__ f4, float* __restrict__ tok) {
  const int bg = blockIdx.x, c = threadIdx.x;            // 384 threads
  float m = -3.0e38f;
  for (int k = 0; k < KK; ++k) m = fmaxf(m, f4[((size_t)bg * KK + k) * DIMC + c]);
  tok[(size_t)bg * DIMC + c] = m;
}

// ------------------- assemble x / pos token sequences ----------------------
__global__ void assemble_kernel(const float* __restrict__ tok, const float* __restrict__ p2,
                                const float* __restrict__ cls_tok, const float* __restrict__ cls_pos,
                                float* __restrict__ x, float* __restrict__ pos) {
  const int bt = blockIdx.x, c = threadIdx.x;            // 384 threads
  const int b = bt / TT, t = bt % TT;
  float xv, pv;
  if (t == 0) { xv = cls_tok[c]; pv = cls_pos[c]; }
  else {
    const size_t r = (size_t)b * GG + (t - 1);
    xv = tok[r * DIMC + c]; pv = p2[r * DIMC + c];
  }
  x[(size_t)bt * DIMC + c] = xv;
  pos[(size_t)bt * DIMC + c] = pv;
}

__global__ void add_inplace_kernel(float* __restrict__ x, const float* __restrict__ y, int n) {
  const int i = blockIdx.x * 256 + threadIdx.x;
  if (i < n) x[i] += y[i];
}

// ------------------------------- LayerNorm ---------------------------------
__global__ void layernorm_kernel(const float* __restrict__ x, const float* __restrict__ g,
                                 const float* __restrict__ b, float* __restrict__ out, int C) {
  const int row = blockIdx.x, tid = threadIdx.x;         // 128 threads
  __shared__ float red[128];
  const float* xr = x + (size_t)row * C;
  float s = 0.0f;
  for (int c = tid; c < C; c += 128) s += xr[c];
  red[tid] = s; __syncthreads();
  for (int st = 64; st > 0; st >>= 1) { if (tid < st) red[tid] += red[tid + st]; __syncthreads(); }
  const float mean = red[0] / (float)C;
  __syncthreads();
  float s2 = 0.0f;
  for (int c = tid; c < C; c += 128) { const float d = xr[c] - mean; s2 += d * d; }
  red[tid] = s2; __syncthreads();
  for (int st = 64; st > 0; st >>= 1) { if (tid < st) red[tid] += red[tid + st]; __syncthreads(); }
  const float inv = rsqrtf(red[0] / (float)C + 1e-5f);
  for (int c = tid; c < C; c += 128)
    out[(size_t)row * C + c] = (xr[c] - mean) * inv * g[c] + b[c];
}

// -------------------------- attention (per b,head) -------------------------
__global__ void attn_kernel(const float* __restrict__ qkv, float* __restrict__ o) {
  const int bh = blockIdx.x, tid = threadIdx.x;          // 256 threads
  const int b = bh / HEADS_, h = bh % HEADS_;
  __shared__ float ks[TT * 64];
  __shared__ float sc[256]; __shared__ float red[256]; __shared__ float qv[64];
  const size_t base = (size_t)b * TT * 1152;
  for (int i = tid; i < TT * 64; i += 256) {
    const int u = i >> 6, dd = i & 63;
    ks[i] = qkv[base + (size_t)u * 1152 + 384 + h * 64 + dd];
  }
  __syncthreads();
  for (int t = 0; t < TT; ++t) {
    if (tid < 64) qv[tid] = qkv[base + (size_t)t * 1152 + h * 64 + tid];
    __syncthreads();
    float sv = -3.0e38f;
    if (tid < TT) {
      float acc = 0.0f;
      for (int dd = 0; dd < 64; ++dd) acc += qv[dd] * ks[tid * 64 + dd];
      sv = acc * 0.125f;                                 // (DIM/HEADS)^-0.5
    }
    red[tid] = sv; __syncthreads();
    for (int st = 128; st > 0; st >>= 1) { if (tid < st) red[tid] = fmaxf(red[tid], red[tid + st]); __syncthreads(); }
    const float mx = red[0];
    __syncthreads();
    const float e = (tid < TT) ? expf(sv - mx) : 0.0f;
    sc[tid] = e; red[tid] = e; __syncthreads();
    for (int st = 128; st > 0; st >>= 1) { if (tid < st) red[tid] += red[tid + st]; __syncthreads(); }
    const float inv = 1.0f / red[0];
    if (tid < 64) {
      float acc = 0.0f;
      for (int u = 0; u < TT; ++u) acc += sc[u] * qkv[base + (size_t)u * 1152 + 768 + h * 64 + tid];
      o[((size_t)b * TT + t) * DIMC + h * 64 + tid] = acc * inv;
    }
    __syncthreads();
  }
}

// -------------------------------- pooling ----------------------------------
__global__ void pool_kernel(const float* __restrict__ xn, float* __restrict__ cf) {
  const int b = blockIdx.x, c = threadIdx.x;             // 384 threads
  cf[(size_t)b * 768 + c] = xn[((size_t)b * TT) * DIMC + c];
  float m = -3.0e38f;
  for (int t = 1; t < TT; ++t) m = fmaxf(m, xn[((size_t)b * TT + t) * DIMC + c]);
  cf[(size_t)b * 768 + 384 + c] = m;
}

// ------------------ small training-mode BN + ReLU (head) -------------------
__global__ void bn_small_relu_kernel(float* __restrict__ x, const float* __restrict__ g,
                                     const float* __restrict__ b, int M, int C) {
  const int c = threadIdx.x;
  if (c >= C) return;
  float s = 0.0f, s2 = 0.0f;
  for (int r = 0; r < M; ++r) { const float v = x[r * C + c]; s += v; s2 += v * v; }
  const float m = s / (float)M;
  const float var = s2 / (float)M - m * m;
  const float scl = g[c] * rsqrtf(var + 1e-5f);
  const float bb = b[c];
  for (int r = 0; r < M; ++r) x[r * C + c] = fmaxf(0.0f, (x[r * C + c] - m) * scl + bb);
}

__global__ void head3_kernel(const float* __restrict__ t2, const float* __restrict__ w,
                             const float* __restrict__ bias, float* __restrict__ out) {
  const int b = blockIdx.x, o = threadIdx.x;             // 64 threads, 40 active
  if (o < 40) {
    float acc = bias[o];
    for (int k = 0; k < 256; ++k) acc += t2[b * 256 + k] * w[o * 256 + k];
    out[b * 40 + o] = acc;
  }
}

// ---------------------------------------------------------------------------
// Host side: parameter mapping + launch sequence
// ---------------------------------------------------------------------------
struct BlkP { const float *ln1g,*ln1b,*qkvw,*projw,*projb,*ln2g,*ln2b,*fc1w,*fc1b,*fc2w,*fc2b; };
struct NetP {
  const float *xyz,*c1w,*c1b,*bn1g,*bn1b,*c2w,*c2b,*c3w,*c3b,*bn2g,*bn2b,*c4w,*c4b,
              *cls_token,*cls_pos,*pw1,*pb1,*pw2,*pb2,*ng,*nb,*h1w,*h1b,*hbn1g,*hbn1b,
              *h2w,*h2b,*hbn2g,*hbn2b,*h3w,*h3b;
  BlkP blk[12];
};

static void map_insertion(void* const* d, NetP& P) {    // recursive dict insertion order
  int i = 0; auto F = [&]() { return (const float*)d[i++]; };
  P.xyz=F(); P.c1w=F(); P.c1b=F(); P.bn1g=F(); P.bn1b=F(); P.c2w=F(); P.c2b=F();
  P.c3w=F(); P.c3b=F(); P.bn2g=F(); P.bn2b=F(); P.c4w=F(); P.c4b=F();
  P.cls_token=F(); P.cls_pos=F(); P.pw1=F(); P.pb1=F(); P.pw2=F(); P.pb2=F();
  for (int l = 0; l < 12; ++l) {
    BlkP& B = P.blk[l];
    B.ln1g=F(); B.ln1b=F(); B.qkvw=F(); B.projw=F(); B.projb=F();
    B.ln2g=F(); B.ln2b=F(); B.fc1w=F(); B.fc1b=F(); B.fc2w=F(); B.fc2b=F();
  }
  P.ng=F(); P.nb=F(); P.h1w=F(); P.h1b=F(); P.hbn1g=F(); P.hbn1b=F();
  P.h2w=F(); P.h2b=F(); P.hbn2g=F(); P.hbn2b=F(); P.h3w=F(); P.h3b=F();
}

static void map_sorted(void* const* d, NetP& P) {       // jax pytree (sorted keys) order
  int i = 0; auto F = [&]() { return (const float*)d[i++]; };
  for (int l = 0; l < 12; ++l) {
    BlkP& B = P.blk[l];
    B.fc1b=F(); B.fc1w=F(); B.fc2b=F(); B.fc2w=F(); B.ln1b=F(); B.ln1g=F();
    B.ln2b=F(); B.ln2g=F(); B.projb=F(); B.projw=F(); B.qkvw=F();
  }
  P.bn1b=F(); P.bn1g=F(); P.bn2b=F(); P.bn2g=F();
  P.c1b=F(); P.c1w=F(); P.c2b=F(); P.c2w=F(); P.c3b=F(); P.c3w=F(); P.c4b=F(); P.c4w=F();
  P.cls_pos=F(); P.cls_token=F();
  P.h1b=F(); P.h1w=F(); P.h2b=F(); P.h2w=F(); P.h3b=F(); P.h3w=F();
  P.hbn1b=F(); P.hbn1g=F(); P.hbn2b=F(); P.hbn2g=F();
  P.nb=F(); P.ng=F(); P.pb1=F(); P.pb2=F(); P.pw1=F(); P.pw2=F();
  P.xyz=F();
}

// workspace layout (float offsets)
static const size_t OFF_CENTER = 0;                        // 12288
static const size_t OFF_NEIGH  = 12288;                    // 393216
static const size_t OFF_STATS  = 405504;                   // 1024
static const size_t OFF_FG     = 406528;                   // 1048576
static const size_t OFF_F1     = 1455104;                  // 16777216 (later f4: 50331648)
static const size_t OFF_F2     = OFF_F1 + (size_t)BGK * 128;   // 33554432
static const size_t OFF_F3     = OFF_F2 + (size_t)BGK * 256;   // 67108864 (later transformer region)

extern "C" void kernel_launch(void* const* d_in, const int* in_sizes, int n_in,
                              void* d_out, int out_size, void* d_ws, size_t ws_size,
                              hipStream_t stream) {
  (void)out_size; (void)ws_size;
  NetP P;
  if (n_in > 0 && in_sizes[0] == BB * NN * 3) map_insertion(d_in, P);
  else                                        map_sorted(d_in, P);

  float* ws     = (float*)d_ws;
  float* center = ws + OFF_CENTER;
  float* neigh  = ws + OFF_NEIGH;
  float* stats  = ws + OFF_STATS;
  float* fg     = ws + OFF_FG;
  float* f1     = ws + OFF_F1;
  float* f2     = ws + OFF_F2;
  float* f3     = ws + OFF_F3;
  float* f4     = f1;                                      // 50331648 = |f1|+|f2|, reuse
  // transformer region overlays f3 after the encoder is done with it
  float* tokens = f3;
  float* p1     = tokens + (size_t)BG * DIMC;
  float* p2     = p1 + (size_t)BG * 128;
  float* x      = p2 + (size_t)BG * DIMC;
  float* posf   = x + (size_t)BT * DIMC;
  float* hbuf   = posf + (size_t)BT * DIMC;
  float* qkvb   = hbuf + (size_t)BT * DIMC;
  float* abuf   = qkvb + (size_t)BT * 1152;
  float* ffb    = abuf + (size_t)BT * DIMC;
  float* xn     = ffb + (size_t)BT * 1536;
  float* cf     = xn + (size_t)BT * DIMC;
  float* t1     = cf + (size_t)BB * 768;
  float* t2     = t1 + 32 * 256;

  // ---- grouping ----
  fps_kernel<<<BB, 256, 0, stream>>>(P.xyz, center);
  knn_kernel<<<BG, 256, 0, stream>>>(P.xyz, center, neigh);

  // ---- PointNet encoder ----
  lin3_kernel<0><<<BGK, 128, 0, stream>>>(neigh, P.c1w, P.c1b, f1);
  hipMemsetAsync(stats, 0, 1024 * sizeof(float), stream);
  colstats_kernel<<<dim3(128, 64), 256, 0, stream>>>(f1, stats, BGK, 128);
  bn_relu_kernel<<<(int)(((size_t)BGK * 128 + 255) / 256), 256, 0, stream>>>(f1, stats, P.bn1g, P.bn1b, BGK, 128);
  gemm_wmma_kernel<0, true, false><<<dim3(BGK / 64, 256 / 64), 128, 0, stream>>>(
      f1, P.c2w, P.c2b, nullptr, f2, BGK, 128, 256, 1, 128);
  group_max_kernel<<<BG, 256, 0, stream>>>(f2, fg);
  // concat GEMM split: broadcast-fg half (W cols 0..255) + f2 half (W cols 256..511)
  gemm_wmma_kernel<0, true, false><<<dim3(BGK / 64, 512 / 64), 128, 0, stream>>>(
      fg, P.c3w, P.c3b, nullptr, f3, BGK, 256, 512, KK, 512);
  gemm_wmma_kernel<0, false, true><<<dim3(BGK / 64, 512 / 64), 128, 0, stream>>>(
      f2, P.c3w + 256, nullptr, f3, f3, BGK, 256, 512, 1, 512);
  hipMemsetAsync(stats, 0, 1024 * sizeof(float), stream);
  colstats_kernel<<<dim3(512, 64), 256, 0, stream>>>(f3, stats, BGK, 512);
  bn_relu_kernel<<<(int)(((size_t)BGK * 512 + 255) / 256), 256, 0, stream>>>(f3, stats, P.bn2g, P.bn2b, BGK, 512);
  gemm_wmma_kernel<0, true, false><<<dim3(BGK / 64, 384 / 64), 128, 0, stream>>>(
      f3, P.c4w, P.c4b, nullptr, f4, BGK, 512, 384, 1, 512);
  token_max_kernel<<<BG, 384, 0, stream>>>(f4, tokens);

  // ---- positional embedding ----
  lin3_kernel<2><<<BG, 128, 0, stream>>>(center, P.pw1, P.pb1, p1);
  gemm_wmma_kernel<0, true, false><<<dim3(BG / 64, 384 / 64), 128, 0, stream>>>(
      p1, P.pw2, P.pb2, nullptr, p2, BG, 128, 384, 1, 128);
  assemble_kernel<<<BT, 384, 0, stream>>>(tokens, p2, P.cls_token, P.cls_pos, x, posf);

  // ---- transformer blocks ----
  const int gx = (BT + 63) / 64;                           // 65 row-tiles
  for (int l = 0; l < 12; ++l) {
    const BlkP& Bp = P.blk[l];
    add_inplace_kernel<<<(BT * DIMC + 255) / 256, 256, 0, stream>>>(x, posf, BT * DIMC);
    layernorm_kernel<<<BT, 128, 0, stream>>>(x, Bp.ln1g, Bp.ln1b, hbuf, DIMC);
    gemm_wmma_kernel<0, false, false><<<dim3(gx, 1152 / 64), 128, 0, stream>>>(
        hbuf, Bp.qkvw, nullptr, nullptr, qkvb, BT, 384, 1152, 1, 384);
    attn_kernel<<<BB * HEADS_, 256, 0, stream>>>(qkvb, abuf);
    gemm_wmma_kernel<0, true, true><<<dim3(gx, 384 / 64), 128, 0, stream>>>(
        abuf, Bp.projw, Bp.projb, x, x, BT, 384, 384, 1, 384);
    layernorm_kernel<<<BT, 128, 0, stream>>>(x, Bp.ln2g, Bp.ln2b, hbuf, DIMC);
    gemm_wmma_kernel<2, true, false><<<dim3(gx, 1536 / 64), 128, 0, stream>>>(
        hbuf, Bp.fc1w, Bp.fc1b, nullptr, ffb, BT, 384, 1536, 1, 384);
    gemm_wmma_kernel<0, true, true><<<dim3(gx, 384 / 64), 128, 0, stream>>>(
        ffb, Bp.fc2w, Bp.fc2b, x, x, BT, 1536, 384, 1, 1536);
  }

  // ---- head ----
  layernorm_kernel<<<BT, 128, 0, stream>>>(x, P.ng, P.nb, xn, DIMC);
  pool_kernel<<<BB, 384, 0, stream>>>(xn, cf);
  gemm_wmma_kernel<0, true, false><<<dim3(1, 4), 128, 0, stream>>>(
      cf, P.h1w, P.h1b, nullptr, t1, 32, 768, 256, 1, 768);
  bn_small_relu_kernel<<<1, 256, 0, stream>>>(t1, P.hbn1g, P.hbn1b, 32, 256);
  gemm_wmma_kernel<0, true, false><<<dim3(1, 4), 128, 0, stream>>>(
      t1, P.h2w, P.h2b, nullptr, t2, 32, 256, 256, 1, 256);
  bn_small_relu_kernel<<<1, 256, 0, stream>>>(t2, P.hbn2g, P.hbn2b, 32, 256);
  head3_kernel<<<BB, 64, 0, stream>>>(t2, P.h3w, P.h3b, (float*)d_out);
}